// SAGE_63780264346292
// MI455X (gfx1250) — compile-verified
//
#include <hip/hip_runtime.h>
#include <hip/hip_bf16.h>

// GCNConv + SAGEConv fused pipeline for MI455X (gfx1250, wave32, WMMA).
// GEMMs use V_WMMA_F32_16X16X4_F32 (full fp32; workload is HBM/L2-bound at
// 23.3 TB/s so low-precision WMMA would not help). W1 is staged into LDS via
// the CDNA5 async global->LDS copy path (ASYNCcnt) so the WMMA loop's B
// fragments come from ds_load instead of competing with the A stream on the
// vector-memory return path.

#define F_IN 256
#define HID  16
#define NCLS 16

typedef __attribute__((ext_vector_type(2))) float v2f;
typedef __attribute__((ext_vector_type(8))) float v8f;

// ---- degree init: self-loop contributes 1 -------------------------------
__global__ void k_init_deg(int* __restrict__ deg, int N) {
    int i = blockIdx.x * blockDim.x + threadIdx.x;
    if (i < N) deg[i] = 1;
}

// ---- in-degree count over edges ----------------------------------------
__global__ void k_count(const int* __restrict__ dst, int E, int* __restrict__ deg) {
    int e = blockIdx.x * blockDim.x + threadIdx.x;
    if (e < E) atomicAdd(&deg[dst[e]], 1);
}

// ---- h = x @ W1 : one 16-row tile per wave, K=256 in steps of 4 --------
__global__ void k_gemm1(const float* __restrict__ x, const float* __restrict__ W1,
                        float* __restrict__ h, int N, int tiles) {
    __shared__ __align__(16) float w1s[F_IN * HID];   // 16 KB
    int tid  = threadIdx.x;
    int lane = tid & 31;
    int wid  = tid >> 5;

    // Stage W1 into LDS with async global->LDS b128 copies (wave 0 only).
    // 32 lanes x 16 B x 32 issues = 16 KB; tracked by ASYNCcnt.
    if (wid == 0) {
        unsigned ldsoff = (unsigned)(uintptr_t)(&w1s[0]) + (unsigned)lane * 16u;
        unsigned goff   = (unsigned)lane * 16u;
        #pragma unroll
        for (int it = 0; it < 32; ++it) {
            asm volatile("global_load_async_to_lds_b128 %0, %1, %2"
                         :: "v"(ldsoff), "v"(goff), "s"(W1) : "memory");
            ldsoff += 512u;
            goff   += 512u;
        }
        asm volatile("s_wait_asynccnt 0" ::: "memory");
    }
    __syncthreads();

    int wave = blockIdx.x * (blockDim.x >> 5) + wid;
    if (wave >= tiles) return;                 // uniform per wave (after barrier)
    int half = lane >> 4;                      // 0: K={0,1}, 1: K={2,3}
    int m    = lane & 15;
    int row  = wave * 16 + m;
    const float* arow = x + (size_t)min(row, N - 1) * F_IN;

    v8f c = {};
    for (int k0 = 0; k0 < F_IN; k0 += 4) {
        // A fragment: row M=m, two consecutive K values (8B-aligned float2)
        v2f a = *(const v2f*)(arow + k0 + 2 * half);
        // B fragment from LDS: W1[k][n], n = m, K split low/high lanes
        v2f b;
        b.x = w1s[(k0 + 2 * half)     * HID + m];
        b.y = w1s[(k0 + 2 * half + 1) * HID + m];
        c = __builtin_amdgcn_wmma_f32_16x16x4_f32(
                false, a, false, b, (short)0, c, false, false);
    }

    // C layout: VGPR j -> row j (lanes 0-15) / row j+8 (lanes 16-31), col m
    if (wave * 16 + 16 <= N) {                 // full tile: straight-line stores
        #pragma unroll
        for (int j = 0; j < 8; ++j)
            h[(size_t)(wave * 16 + j + 8 * half) * HID + m] = c[j];
    } else {
        #pragma unroll
        for (int j = 0; j < 8; ++j) {
            int r = wave * 16 + j + 8 * half;
            if (r < N) h[(size_t)r * HID + m] = c[j];
        }
    }
}

// ---- dinv = rsqrt(deg); h1 = self-loop term + b1; zero agg -------------
__global__ void k_prep(const int* __restrict__ deg, const float* __restrict__ h,
                       const float* __restrict__ b1, float* __restrict__ dinv,
                       float* __restrict__ h1, float* __restrict__ agg, int N) {
    int idx = blockIdx.x * blockDim.x + threadIdx.x;
    if (idx >= N * HID) return;
    int node = idx >> 4, f = idx & 15;
    float di = rsqrtf((float)deg[node]);       // deg >= 1 always (self-loop)
    if (f == 0) dinv[node] = di;
    h1[idx]  = h[idx] * di * di + b1[f];       // self-loop norm = dinv^2
    agg[idx] = 0.0f;
}

// ---- GCN edge scatter: h1[dst] += h[src] * dinv[src]*dinv[dst] ---------
__global__ void k_scatter_gcn(const int* __restrict__ src, const int* __restrict__ dst,
                              const float* __restrict__ h, const float* __restrict__ dinv,
                              float* __restrict__ h1, int total) {
    int idx = blockIdx.x * blockDim.x + threadIdx.x;
    if (idx >= total) return;
    int e = idx >> 4, f = idx & 15;            // 16 lanes per edge, coalesced rows
    int s = src[e], d = dst[e];
    float w = dinv[s] * dinv[d];
    atomicAdd(&h1[(size_t)d * HID + f], h[(size_t)s * HID + f] * w);
}

// ---- SAGE edge scatter: agg[dst] += h1[src] ----------------------------
__global__ void k_scatter_sage(const int* __restrict__ src, const int* __restrict__ dst,
                               const float* __restrict__ h1, float* __restrict__ agg,
                               int total) {
    int idx = blockIdx.x * blockDim.x + threadIdx.x;
    if (idx >= total) return;
    int e = idx >> 4, f = idx & 15;
    atomicAdd(&agg[(size_t)dst[e] * HID + f], h1[(size_t)src[e] * HID + f]);
}

// ---- out = (agg/cnt)@Wl + h1@Wr + b2, then log_softmax -----------------
__global__ void k_sage_out(const float* __restrict__ h1, const float* __restrict__ agg,
                           const int* __restrict__ deg,
                           const float* __restrict__ Wl, const float* __restrict__ Wr,
                           const float* __restrict__ b2,
                           float* __restrict__ out, int N, int tiles) {
    int wave = blockIdx.x * (blockDim.x >> 5) + (threadIdx.x >> 5);
    if (wave >= tiles) return;
    int lane = threadIdx.x & 31;
    int half = lane >> 4;
    int m    = lane & 15;
    int row  = min(wave * 16 + m, N - 1);
    float ic = 1.0f / fmaxf((float)(deg[row] - 1), 1.0f);   // mean divisor
    const float* ar = agg + (size_t)row * HID;
    const float* hr = h1  + (size_t)row * HID;

    v8f c = {};
#pragma unroll
    for (int k0 = 0; k0 < HID; k0 += 4) {      // (agg/cnt) @ Wl
        v2f a = *(const v2f*)(ar + k0 + 2 * half);
        a.x *= ic; a.y *= ic;
        v2f b;
        b.x = Wl[(k0 + 2 * half)     * NCLS + m];
        b.y = Wl[(k0 + 2 * half + 1) * NCLS + m];
        c = __builtin_amdgcn_wmma_f32_16x16x4_f32(
                false, a, false, b, (short)0, c, false, false);
    }
#pragma unroll
    for (int k0 = 0; k0 < HID; k0 += 4) {      // + h1 @ Wr (accumulate in C)
        v2f a = *(const v2f*)(hr + k0 + 2 * half);
        v2f b;
        b.x = Wr[(k0 + 2 * half)     * NCLS + m];
        b.y = Wr[(k0 + 2 * half + 1) * NCLS + m];
        c = __builtin_amdgcn_wmma_f32_16x16x4_f32(
                false, a, false, b, (short)0, c, false, false);
    }

    float bv = b2[m];
    bool full = (wave * 16 + 16 <= N);
#pragma unroll
    for (int j = 0; j < 8; ++j) {
        float v  = c[j] + bv;
        // classes of one output row live in one 16-lane group -> xor reduce
        float mx = v;
        for (int off = 1; off < 16; off <<= 1)
            mx = fmaxf(mx, __shfl_xor(mx, off, 32));
        float s = __expf(v - mx);
        for (int off = 1; off < 16; off <<= 1)
            s += __shfl_xor(s, off, 32);
        int r = wave * 16 + j + 8 * half;
        if (full || r < N) out[(size_t)r * NCLS + m] = v - mx - __logf(s);
    }
}

extern "C" void kernel_launch(void* const* d_in, const int* in_sizes, int n_in,
                              void* d_out, int out_size, void* d_ws, size_t ws_size,
                              hipStream_t stream) {
    const float* x    = (const float*)d_in[0];
    const int*   edge = (const int*)  d_in[1];
    const float* W1   = (const float*)d_in[2];
    const float* b1   = (const float*)d_in[3];
    const float* Wl   = (const float*)d_in[4];
    const float* Wr   = (const float*)d_in[5];
    const float* b2   = (const float*)d_in[6];
    float*       out  = (float*)d_out;

    int N = in_sizes[0] / F_IN;
    int E = in_sizes[1] / 2;
    const int* src = edge;
    const int* dst = edge + E;

    // Workspace layout (all 8B-aligned for float2 fragment loads)
    char* ws = (char*)d_ws;
    int*   deg  = (int*)  ws;  ws += (size_t)N * sizeof(int);
    float* dinv = (float*)ws;  ws += (size_t)N * sizeof(float);
    float* h    = (float*)ws;  ws += (size_t)N * HID * sizeof(float);
    float* h1   = (float*)ws;  ws += (size_t)N * HID * sizeof(float);
    float* agg  = (float*)ws;

    int tiles = (N + 15) / 16;
    int totNF = N * HID;
    int totEF = E * HID;

    k_init_deg    <<<(N + 255) / 256,     256, 0, stream>>>(deg, N);
    k_count       <<<(E + 255) / 256,     256, 0, stream>>>(dst, E, deg);
    k_gemm1       <<<(tiles + 7) / 8,     256, 0, stream>>>(x, W1, h, N, tiles);
    k_prep        <<<(totNF + 255) / 256, 256, 0, stream>>>(deg, h, b1, dinv, h1, agg, N);
    k_scatter_gcn <<<(totEF + 255) / 256, 256, 0, stream>>>(src, dst, h, dinv, h1, totEF);
    k_scatter_sage<<<(totEF + 255) / 256, 256, 0, stream>>>(src, dst, h1, agg, totEF);
    k_sage_out    <<<(tiles + 7) / 8,     256, 0, stream>>>(h1, agg, deg, Wl, Wr, b2, out, N, tiles);
}